// GAT_489626272249
// MI455X (gfx1250) — compile-verified
//
#include <hip/hip_runtime.h>

#define N_NODES 100000
#define N_EDGES 1600000
#define EE      (N_EDGES + N_NODES)   // edges + self loops = 1,700,000
#define C       64
#define H       4
#define HC      (H * C)               // 256
#define G_DRONES 64
#define NODE_F  32
#define OUT_F   32

typedef float v2f __attribute__((ext_vector_type(2)));
typedef float v8f __attribute__((ext_vector_type(8)));

// ---------------------------------------------------------------------------
// fp32 WMMA helper: D(16x16) += A(16x4) * B(4x16), wave32.
// A layout: lanes 0-15 hold row M=lane (K=k0,k0+1); lanes 16-31 hold K=k0+2,k0+3.
// B layout: symmetric over columns. C/D: vgpr r -> (M=r | r+8, N=lane%16).
// ---------------------------------------------------------------------------
__device__ inline v8f wmma_f32_k4(v2f a, v2f b, v8f c) {
    return __builtin_amdgcn_wmma_f32_16x16x4_f32(
        /*neg_a=*/false, a, /*neg_b=*/false, b,
        /*c_mod=*/(short)0, c, /*reuse_a=*/false, /*reuse_b=*/false);
}

// aligned 8-byte vector load -> single global_load_b64
__device__ inline v2f ld2(const float* __restrict__ p) {
    return *(const v2f*)p;
}

// float atomic max via single monotonic int atomic (no CAS loop)
__device__ inline void atomicMaxF(float* addr, float v) {
    int iv = __float_as_int(v);
    if (iv >= 0) atomicMax((int*)addr, iv);
    else         atomicMin((unsigned int*)addr, (unsigned int)iv);
}

// ---------------------------------------------------------------------------
// drone_out[g,c] = drone_feat[g,:] . drone_W[c,:] + drone_b[c]
// ---------------------------------------------------------------------------
__global__ void k_drone(const float* __restrict__ df, const float* __restrict__ dW,
                        const float* __restrict__ db, float* __restrict__ dout) {
    int gid = blockIdx.x * blockDim.x + threadIdx.x;
    if (gid >= G_DRONES * C) return;
    int g = gid >> 6, c = gid & 63;
    float acc = db[c];
#pragma unroll
    for (int f = 0; f < 16; ++f) acc += df[g * 16 + f] * dW[c * 16 + f];
    dout[gid] = acc;
}

// ---------------------------------------------------------------------------
// h[n,c] = x[n,:] @ node_W[c,:] + node_b[c] + drone_out[batch[n],c]   (WMMA)
// ---------------------------------------------------------------------------
__global__ void k_embed(const float* __restrict__ x, const float* __restrict__ nodeW,
                        const float* __restrict__ node_b, const float* __restrict__ drone_out,
                        const int* __restrict__ batch, float* __restrict__ h) {
    int wave = (blockIdx.x * blockDim.x + threadIdx.x) >> 5;
    int lane = threadIdx.x & 31;
    const int ntiles = C / 16;                 // 4 col-tiles
    int mt = wave / ntiles, nt = wave % ntiles;
    if (mt >= N_NODES / 16) return;            // wave-uniform guard
    int m0 = mt * 16, n0 = nt * 16;
    int hi = lane >> 4, lr = lane & 15;
    v8f acc = {};
#pragma unroll
    for (int k0 = 0; k0 < NODE_F; k0 += 4) {
        int ka = k0 + 2 * hi;
        v2f a = ld2(x + (m0 + lr) * NODE_F + ka);
        v2f b = ld2(nodeW + (n0 + lr) * NODE_F + ka);
        acc = wmma_f32_k4(a, b, acc);
    }
    int col = n0 + lr;
    float nb = node_b[col];
#pragma unroll
    for (int r = 0; r < 8; ++r) {
        int row = m0 + r + 8 * hi;
        int g = batch[row];
        h[row * C + col] = acc[r] + nb + drone_out[g * C + col];
    }
}

// ---------------------------------------------------------------------------
// xh[n, 0:256] = h[n,:] @ convW[j,:]^T      (WMMA, K=64 -> 16 k-steps)
// ---------------------------------------------------------------------------
__global__ void k_gemm_xh(const float* __restrict__ h, const float* __restrict__ W,
                          float* __restrict__ xh) {
    int wave = (blockIdx.x * blockDim.x + threadIdx.x) >> 5;
    int lane = threadIdx.x & 31;
    const int ntiles = HC / 16;                // 16 col-tiles
    int mt = wave / ntiles, nt = wave % ntiles;
    if (mt >= N_NODES / 16) return;
    int m0 = mt * 16, n0 = nt * 16;
    int hi = lane >> 4, lr = lane & 15;
    v8f acc = {};
#pragma unroll
    for (int k0 = 0; k0 < C; k0 += 4) {
        int ka = k0 + 2 * hi;
        v2f a = ld2(h + (m0 + lr) * C + ka);
        v2f b = ld2(W + (n0 + lr) * C + ka);
        acc = wmma_f32_k4(a, b, acc);
    }
    int col = n0 + lr;
#pragma unroll
    for (int r = 0; r < 8; ++r) xh[(m0 + r + 8 * hi) * HC + col] = acc[r];
}

// ---------------------------------------------------------------------------
// alpha_s[n,h] = xh[n,h,:] . att_src[h,:] ; alpha_d likewise
// ---------------------------------------------------------------------------
__global__ void k_alpha(const float* __restrict__ xh, const float* __restrict__ a_src,
                        const float* __restrict__ a_dst, float* __restrict__ alpha_s,
                        float* __restrict__ alpha_d) {
    int gid = blockIdx.x * blockDim.x + threadIdx.x;
    if (gid >= N_NODES * H) return;
    int n = gid >> 2, hh = gid & 3;
    const float4* xv = (const float4*)(xh + (size_t)n * HC + hh * C);
    const float4* sv = (const float4*)(a_src + hh * C);
    const float4* dv = (const float4*)(a_dst + hh * C);
    float ss = 0.f, dd = 0.f;
#pragma unroll
    for (int i = 0; i < 16; ++i) {
        float4 xx = xv[i], aa = sv[i], bb = dv[i];
        ss += xx.x * aa.x + xx.y * aa.y + xx.z * aa.z + xx.w * aa.w;
        dd += xx.x * bb.x + xx.y * bb.y + xx.z * bb.z + xx.w * bb.w;
    }
    alpha_s[gid] = ss;
    alpha_d[gid] = dd;
}

// acc = 0, m = -inf, s = 0
__global__ void k_init(float* __restrict__ acc, float* __restrict__ m, float* __restrict__ s) {
    int gid = blockIdx.x * blockDim.x + threadIdx.x;
    if (gid < N_NODES * HC) acc[gid] = 0.f;
    if (gid < N_NODES * H) { m[gid] = -__builtin_inff(); s[gid] = 0.f; }
}

// ---------------------------------------------------------------------------
// pass 1: segment max of leaky_relu(alpha_s[src]+alpha_d[dst]) over dst
// ---------------------------------------------------------------------------
__global__ void k_edge_max(const int* __restrict__ ei, const float* __restrict__ as,
                           const float* __restrict__ ad, float* __restrict__ m) {
    int gid = blockIdx.x * blockDim.x + threadIdx.x;
    if (gid >= EE * H) return;
    int e = gid >> 2, hh = gid & 3;
    int src, dst;
    if (e < N_EDGES) { src = ei[e]; dst = ei[N_EDGES + e]; }
    else             { src = dst = e - N_EDGES; }          // self loop
    float a  = as[src * H + hh] + ad[dst * H + hh];
    float ev = a > 0.f ? a : 0.2f * a;
    atomicMaxF(&m[dst * H + hh], ev);
}

// ---------------------------------------------------------------------------
// pass 2 (fused): ex = exp(e - m[dst]); s[dst] += ex; acc[dst] += ex*xh[src]
// 8 threads/edge: (head, 32-channel half), float4 gathers from L2-resident xh
// ---------------------------------------------------------------------------
__global__ void k_edge_acc(const int* __restrict__ ei, const float* __restrict__ as,
                           const float* __restrict__ ad, const float* __restrict__ m,
                           const float* __restrict__ xh, float* __restrict__ s,
                           float* __restrict__ acc) {
    int gid = blockIdx.x * blockDim.x + threadIdx.x;
    if (gid >= EE * 8) return;
    int e = gid >> 3, p = gid & 7;
    int hh = p >> 1, c0 = (p & 1) * 32;
    int src, dst;
    if (e < N_EDGES) { src = ei[e]; dst = ei[N_EDGES + e]; }
    else             { src = dst = e - N_EDGES; }
    float a  = as[src * H + hh] + ad[dst * H + hh];
    float ev = a > 0.f ? a : 0.2f * a;
    float ex = __expf(ev - m[dst * H + hh]);
    if ((p & 1) == 0) atomicAdd(&s[dst * H + hh], ex);
    const float4* xv = (const float4*)(xh + (size_t)src * HC + hh * C + c0);
    float* ao = acc + (size_t)dst * HC + hh * C + c0;
#pragma unroll
    for (int i = 0; i < 8; ++i) {
        float4 v = xv[i];
        atomicAdd(ao + 4 * i + 0, ex * v.x);
        atomicAdd(ao + 4 * i + 1, ex * v.y);
        atomicAdd(ao + 4 * i + 2, ex * v.z);
        atomicAdd(ao + 4 * i + 3, ex * v.w);
    }
}

// ---------------------------------------------------------------------------
// per-node: normalize by s, head-mean + bias, LayerNorm, ReLU, residual add.
// One wave32 per node, 2 channels per lane, __shfl_xor reductions.
// ---------------------------------------------------------------------------
__global__ void k_node_update(const float* __restrict__ acc, const float* __restrict__ s,
                              const float* __restrict__ cb, const float* __restrict__ g,
                              const float* __restrict__ b, float* __restrict__ h) {
    int gid = blockIdx.x * blockDim.x + threadIdx.x;
    int n = gid >> 5, lane = gid & 31;
    if (n >= N_NODES) return;
    float inv[H];
#pragma unroll
    for (int hh = 0; hh < H; ++hh) inv[hh] = 1.f / (s[n * H + hh] + 1e-16f);
    int c1 = lane, c2 = lane + 32;
    float v1 = 0.f, v2 = 0.f;
#pragma unroll
    for (int hh = 0; hh < H; ++hh) {
        v1 += acc[(size_t)n * HC + hh * C + c1] * inv[hh];
        v2 += acc[(size_t)n * HC + hh * C + c2] * inv[hh];
    }
    v1 = 0.25f * v1 + cb[c1];
    v2 = 0.25f * v2 + cb[c2];
    float sum = v1 + v2;
#pragma unroll
    for (int o = 16; o > 0; o >>= 1) sum += __shfl_xor(sum, o, 32);
    float mu = sum * (1.f / 64.f);
    float d1 = v1 - mu, d2 = v2 - mu;
    float vs = d1 * d1 + d2 * d2;
#pragma unroll
    for (int o = 16; o > 0; o >>= 1) vs += __shfl_xor(vs, o, 32);
    float rstd = rsqrtf(vs * (1.f / 64.f) + 1e-5f);
    float y1 = d1 * rstd * g[c1] + b[c1]; y1 = y1 > 0.f ? y1 : 0.f;
    float y2 = d2 * rstd * g[c2] + b[c2]; y2 = y2 > 0.f ? y2 : 0.f;
    h[n * C + c1] += y1;
    h[n * C + c2] += y2;
}

// ---------------------------------------------------------------------------
// out[n, 0:32] = h[n,:] @ out_W^T + out_b      (WMMA)
// ---------------------------------------------------------------------------
__global__ void k_out(const float* __restrict__ h, const float* __restrict__ Wo,
                      const float* __restrict__ bo, float* __restrict__ out) {
    int wave = (blockIdx.x * blockDim.x + threadIdx.x) >> 5;
    int lane = threadIdx.x & 31;
    const int ntiles = OUT_F / 16;             // 2
    int mt = wave / ntiles, nt = wave % ntiles;
    if (mt >= N_NODES / 16) return;
    int m0 = mt * 16, n0 = nt * 16;
    int hi = lane >> 4, lr = lane & 15;
    v8f acc = {};
#pragma unroll
    for (int k0 = 0; k0 < C; k0 += 4) {
        int ka = k0 + 2 * hi;
        v2f a = ld2(h + (m0 + lr) * C + ka);
        v2f b = ld2(Wo + (n0 + lr) * C + ka);
        acc = wmma_f32_k4(a, b, acc);
    }
    int col = n0 + lr;
    float bb = bo[col];
#pragma unroll
    for (int r = 0; r < 8; ++r) out[(m0 + r + 8 * hi) * OUT_F + col] = acc[r] + bb;
}

extern "C" void kernel_launch(void* const* d_in, const int* in_sizes, int n_in,
                              void* d_out, int out_size, void* d_ws, size_t ws_size,
                              hipStream_t stream) {
    const float* x          = (const float*)d_in[0];
    const float* drone_feat = (const float*)d_in[1];
    const int*   edge_index = (const int*)d_in[2];
    const int*   batch      = (const int*)d_in[3];
    const float* node_W     = (const float*)d_in[4];
    const float* node_b     = (const float*)d_in[5];
    const float* drone_W    = (const float*)d_in[6];
    const float* drone_b    = (const float*)d_in[7];
    const float* convW[2]   = {(const float*)d_in[8],  (const float*)d_in[14]};
    const float* att_src[2] = {(const float*)d_in[9],  (const float*)d_in[15]};
    const float* att_dst[2] = {(const float*)d_in[10], (const float*)d_in[16]};
    const float* convb[2]   = {(const float*)d_in[11], (const float*)d_in[17]};
    const float* ln_g[2]    = {(const float*)d_in[12], (const float*)d_in[18]};
    const float* ln_b[2]    = {(const float*)d_in[13], (const float*)d_in[19]};
    const float* out_W      = (const float*)d_in[20];
    const float* out_b      = (const float*)d_in[21];

    // workspace layout (floats): ~236.8 MB total
    float* ws = (float*)d_ws;
    size_t off = 0;
    float* h        = ws + off; off += (size_t)N_NODES * C;   // 25.6 MB
    float* xh       = ws + off; off += (size_t)N_NODES * HC;  // 102.4 MB (fits L2)
    float* acc      = ws + off; off += (size_t)N_NODES * HC;  // 102.4 MB
    float* alpha_s  = ws + off; off += (size_t)N_NODES * H;
    float* alpha_d  = ws + off; off += (size_t)N_NODES * H;
    float* mbuf     = ws + off; off += (size_t)N_NODES * H;
    float* sbuf     = ws + off; off += (size_t)N_NODES * H;
    float* drone_o  = ws + off; off += (size_t)G_DRONES * C;

    dim3 blk(256);
    k_drone<<<(G_DRONES * C + 255) / 256, blk, 0, stream>>>(drone_feat, drone_W, drone_b, drone_o);
    k_embed<<<(N_NODES / 16) * (C / 16) / 8, blk, 0, stream>>>(x, node_W, node_b, drone_o, batch, h);

    for (int l = 0; l < 2; ++l) {
        k_init<<<(N_NODES * HC + 255) / 256, blk, 0, stream>>>(acc, mbuf, sbuf);
        k_gemm_xh<<<(N_NODES / 16) * (HC / 16) / 8, blk, 0, stream>>>(h, convW[l], xh);
        k_alpha<<<(N_NODES * H + 255) / 256, blk, 0, stream>>>(xh, att_src[l], att_dst[l],
                                                               alpha_s, alpha_d);
        k_edge_max<<<(EE * H + 255) / 256, blk, 0, stream>>>(edge_index, alpha_s, alpha_d, mbuf);
        k_edge_acc<<<(EE * 8 + 255) / 256, blk, 0, stream>>>(edge_index, alpha_s, alpha_d, mbuf,
                                                             xh, sbuf, acc);
        k_node_update<<<(N_NODES * 32 + 255) / 256, blk, 0, stream>>>(acc, sbuf, convb[l],
                                                                      ln_g[l], ln_b[l], h);
    }

    int out_tiles = (N_NODES / 16) * (OUT_F / 16);  // 12500
    k_out<<<(out_tiles + 7) / 8, blk, 0, stream>>>(h, out_W, out_b, (float*)d_out);
}